// SoftmaxCategoricalHead_44650480009270
// MI455X (gfx1250) — compile-verified
//
#include <hip/hip_runtime.h>
#include <stdint.h>

// Sampling head: repetition penalty -> /temperature -> top-k(50) -> top-p(0.9) -> softmax
// BATCH=256 rows, VOCAB=128000, HIST=200 prev tokens per row.
// One 256-thread block per row. Logits (131MB) stay L2-resident across passes
// (MI455X L2 = 192MB); output written with non-temporal B128 stores so the
// probs stream does not evict the logits between passes.

#define BATCH_ 256
#define VOCAB_ 128000
#define HIST_  200
#define TOPK_  50
#define NT_    256
#define CAP_   256

typedef __attribute__((address_space(1))) int gint_t;   // global (AS1) int
typedef __attribute__((address_space(3))) int lint_t;   // LDS (AS3) int
typedef float v4f __attribute__((ext_vector_type(4)));

__device__ __forceinline__ uint32_t f2key(float f) {
  // monotone float -> uint key (larger float => larger key)
  uint32_t u = __float_as_uint(f);
  return (u & 0x80000000u) ? ~u : (u | 0x80000000u);
}

__global__ __launch_bounds__(NT_) void sampling_head_kernel(
    const float* __restrict__ logits, const int* __restrict__ prev,
    float* __restrict__ out) {
  __shared__ uint32_t mask_[4096];   // 16KB: vocab membership bitmask (4000 words used)
  __shared__ uint32_t hist_[4096];   // 16KB: 12-bit key histogram
  __shared__ uint32_t part_[NT_];
  __shared__ int      ptok_[HIST_];
  __shared__ float    cand_[CAP_];
  __shared__ float    sort_[CAP_];
  __shared__ uint32_t candCount_, foundFlag_, selBin_;
  __shared__ float    maxS_, cutS_, invSfS_;

  const int row = blockIdx.x;
  const int t   = threadIdx.x;
  const float* rl = logits + (size_t)row * VOCAB_;
  float*       ro = out    + (size_t)row * VOCAB_;

  // ---- Stage prev_tokens into LDS via gfx1250 async global->LDS (ASYNCcnt) ----
  if (t < HIST_) {
    const int* g = prev + (size_t)row * HIST_ + t;
    __builtin_amdgcn_global_load_async_to_lds_b32(
        (gint_t*)(uintptr_t)g,
        (lint_t*)(uintptr_t)(uint32_t)(uintptr_t)&ptok_[t],
        0, 0);
  }
  for (int i = t; i < 4096; i += NT_) mask_[i] = 0u;
  if (t == 0) candCount_ = 0u;
  __builtin_amdgcn_s_wait_asynccnt(0);  // each wave waits for its own async copies
  __syncthreads();
  if (t < HIST_) {
    uint32_t tok = (uint32_t)ptok_[t];
    atomicOr(&mask_[tok >> 5], 1u << (tok & 31u));
  }
  __syncthreads();

  // Vectorized modified logit: one B128 load + one LDS mask word per 4 elements.
  // (c is a multiple of 4, so all 4 indices share mask word c>>5.)
  auto modv4 = [&](int c) -> v4f {
    v4f x = *(const v4f*)(rl + c);
    uint32_t mw = mask_[c >> 5];
    uint32_t sh = (uint32_t)(c & 31);
#pragma unroll
    for (int j = 0; j < 4; ++j) {
      if ((mw >> (sh + j)) & 1u) x[j] = (x[j] < 0.0f) ? x[j] * 1.2f : x[j] / 1.2f;
      x[j] = x[j] / 0.6f;
    }
    return x;
  };

  // ---- Pass 1: 4096-bin histogram on key[31:20]; find bin holding rank TOPK_ ----
  // Attempt 0 histograms only non-negative values (key >= 0x80000000) to halve
  // LDS-atomic traffic; attempt 1 (fallback, never expected here) counts all.
  for (int attempt = 0; attempt < 2; ++attempt) {
    for (int i = t; i < 4096; i += NT_) hist_[i] = 0u;
    __syncthreads();
    const uint32_t lim = (attempt == 0) ? (2048u << 20) : 0u;
    for (int c = t * 4; c < VOCAB_; c += NT_ * 4) {
      __builtin_prefetch(rl + c + NT_ * 16, 0, 0);  // global_prefetch_b8
      v4f x = modv4(c);
#pragma unroll
      for (int j = 0; j < 4; ++j) {
        uint32_t key = f2key(x[j]);
        if (key >= lim) atomicAdd(&hist_[key >> 20], 1u);
      }
    }
    __syncthreads();
    uint32_t s = 0;
    for (int j = 0; j < 16; ++j) s += hist_[t * 16 + j];
    part_[t] = s;
    __syncthreads();
    if (t == 0) {
      uint32_t cum = 0;
      int seg, found = 0;
      for (seg = 255; seg >= 0; --seg) {
        if (cum + part_[seg] >= (uint32_t)TOPK_) { found = 1; break; }
        cum += part_[seg];
      }
      if (found) {
        int b = seg * 16 + 15;
        while (cum + hist_[b] < (uint32_t)TOPK_) { cum += hist_[b]; --b; }
        selBin_ = (uint32_t)b;
      }
      foundFlag_ = (uint32_t)found;
    }
    __syncthreads();
    if (foundFlag_) break;
  }

  // ---- Pass 2: collect all values with key >= lower bound of selected bin ----
  const uint32_t keyLow = selBin_ << 20;
  for (int c = t * 4; c < VOCAB_; c += NT_ * 4) {
    v4f x = modv4(c);
#pragma unroll
    for (int j = 0; j < 4; ++j) {
      if (f2key(x[j]) >= keyLow) {
        uint32_t idx = atomicAdd(&candCount_, 1u);
        if (idx < CAP_) cand_[idx] = x[j];
      }
    }
  }
  __syncthreads();
  const int n = (candCount_ < (uint32_t)CAP_) ? (int)candCount_ : CAP_;

  // ---- Rank-sort candidates descending (n <= 256, one per thread) ----
  if (t < n) {
    float my = cand_[t];
    int r = 0;
    for (int j = 0; j < n; ++j) {
      float vj = cand_[j];
      r += (int)((vj > my) || (vj == my && j < t));
    }
    sort_[r] = my;
  }
  __syncthreads();

  // ---- Top-k kth value, top-p cutoff, renormalization (thread 0; n is tiny) ----
  if (t == 0) {
    float mx  = sort_[0];
    float kth = sort_[TOPK_ - 1];           // 50th largest; ties beyond it also kept
    int nK = TOPK_;
    while (nK < n && sort_[nK] == kth) ++nK;
    // softmax over top-k survivors (everything else is -inf => prob 0)
    float S = 0.f;
    for (int i = 0; i < nK; ++i) { float e = expf(sort_[i] - mx); cand_[i] = e; S += e; }
    float invS = 1.f / S;
    // HF shifted-cumsum: token i removed iff cumprob of tokens 0..i-1 > 0.9
    float cum = 0.f;
    int m = nK;
    for (int i = 0; i < nK; ++i) {
      if (cum > 0.9f) { m = i; break; }
      cum += cand_[i] * invS;
    }
    float Sf = 0.f;
    for (int i = 0; i < m; ++i) Sf += cand_[i];
    maxS_   = mx;
    cutS_   = sort_[m - 1];                 // smallest kept modified-logit value
    invSfS_ = 1.f / Sf;
  }
  __syncthreads();

  // ---- Pass 3: emit final probs; NT B128 stores keep logits resident in L2 ----
  const float mx = maxS_, cut = cutS_, isf = invSfS_;
  for (int c = t * 4; c < VOCAB_; c += NT_ * 4) {
    v4f x = modv4(c);
    v4f p;
#pragma unroll
    for (int j = 0; j < 4; ++j) {
      float pj = 0.f;
      if (x[j] >= cut) pj = expf(x[j] - mx) * isf;
      p[j] = pj;
    }
    __builtin_nontemporal_store(p, (v4f*)(ro + c));
  }
}

extern "C" void kernel_launch(void* const* d_in, const int* in_sizes, int n_in,
                              void* d_out, int out_size, void* d_ws, size_t ws_size,
                              hipStream_t stream) {
  (void)in_sizes; (void)n_in; (void)out_size; (void)d_ws; (void)ws_size;
  const float* logits = (const float*)d_in[0];
  const int*   prev   = (const int*)d_in[1];
  float*       out    = (float*)d_out;
  hipLaunchKernelGGL(sampling_head_kernel, dim3(BATCH_), dim3(NT_), 0, stream,
                     logits, prev, out);
}